// GRI_5488968204831
// MI455X (gfx1250) — compile-verified
//
#include <hip/hip_runtime.h>
#include <math.h>

#define BATCH   2048
#define NNODES  4096
#define LATENT  128
#define KFAN    16
#define DROW    (1 + LATENT + KFAN)   // 145

typedef __attribute__((ext_vector_type(2))) float v2f;
typedef __attribute__((ext_vector_type(8))) float v8f;

// ---------------------------------------------------------------------------
// Kernel 1: pre[b,n] = w[n,0] + sum_l z[b,l]*w[n,1+l]   via V_WMMA_F32_16X16X4_F32
// One wave per 16x16 (batch x node) tile, K=128 in 32 steps of 4.
// Result stored NODE-MAJOR into ws: ws[n*BATCH + b]  (so the DAG sweep gathers
// coalesced lines), later overwritten in place by the activation.
// ---------------------------------------------------------------------------
__global__ __launch_bounds__(256)
void wmma_pre_kernel(const float* __restrict__ z,
                     const float* __restrict__ w,
                     float* __restrict__ ws) {
    const int wave = blockIdx.x * 8 + (threadIdx.x >> 5);
    const int lane = threadIdx.x & 31;
    const int bt   = wave & 127;          // batch tile  (128 tiles of 16)
    const int nt   = wave >> 7;           // node tile   (256 tiles of 16)
    const int half = lane >> 4;           // 0: K=0,1  1: K=2,3
    const int m    = lane & 15;           // M for A, N for B/C/D

    const float* zrow = z + (bt * 16 + m) * LATENT + 2 * half;
    const float* wrow = w + (nt * 16 + m) * DROW + 1 + 2 * half;

    v8f acc = {};
#pragma unroll 8
    for (int k = 0; k < LATENT; k += 4) {
        v2f a, b;
        a.x = zrow[k];     a.y = zrow[k + 1];
        b.x = wrow[k];     b.y = wrow[k + 1];
        acc = __builtin_amdgcn_wmma_f32_16x16x4_f32(
            /*neg_a=*/false, a, /*neg_b=*/false, b,
            /*c_mod=*/(short)0, acc, /*reuse_a=*/false, /*reuse_b=*/false);
    }

    const float bias = w[(nt * 16 + m) * DROW];   // per-N bias, uniform over M

    // D element (M,N): M = r + 8*half, N = m.  Node-major store:
    // addr = n*BATCH + b = (nt*16+m)*BATCH + bt*16 + 8*half + r  -> 8 consecutive
    float* dst = ws + (nt * 16 + m) * BATCH + bt * 16 + 8 * half;
    float4 lo = make_float4(acc[0] + bias, acc[1] + bias, acc[2] + bias, acc[3] + bias);
    float4 hi = make_float4(acc[4] + bias, acc[5] + bias, acc[6] + bias, acc[7] + bias);
    *reinterpret_cast<float4*>(dst)     = lo;
    *reinterpret_cast<float4*>(dst + 4) = hi;
}

// ---------------------------------------------------------------------------
// Kernel 2: sequential DAG sweep. One thread per batch row; all dependencies
// are within-thread (store->load same address, same wave: hardware-ordered).
// Parent indices + parent weights are wave-uniform -> scalar loads; the 16
// gathers per node are independent coalesced vector loads (pipelined).
// ws slot i*BATCH+b holds pre value, overwritten with tanh(activation).
// ---------------------------------------------------------------------------
__global__ __launch_bounds__(256)
void dag_sweep_kernel(const float* __restrict__ w,
                      const int*  __restrict__ parents,
                      float* __restrict__ ws) {
    const int b = blockIdx.x * blockDim.x + threadIdx.x;   // 0..2047
    if (b >= BATCH) return;

    for (int i = 0; i < NNODES; ++i) {
        float acc = ws[i * BATCH + b];                      // pre-activation
        const int*   par = parents + i * KFAN;              // wave-uniform
        const float* pw  = w + i * DROW + 1 + LATENT;       // wave-uniform
#pragma unroll
        for (int k = 0; k < KFAN; ++k) {
            // padded slots have pw[k]==0 (weights pre-masked in setup)
            acc = fmaf(pw[k], ws[par[k] * BATCH + b], acc);
        }
        ws[i * BATCH + b] = tanhf(acc);
    }
}

// ---------------------------------------------------------------------------
// Kernel 3: node-major -> (B,N) output, LDS 32x32 tile transpose,
// coalesced on both sides.
// ---------------------------------------------------------------------------
__global__ __launch_bounds__(256)
void transpose_kernel(const float* __restrict__ ws, float* __restrict__ out) {
    __shared__ float tile[32][33];
    const int tx = threadIdx.x;           // 0..31
    const int ty = threadIdx.y;           // 0..7
    const int bBase = blockIdx.x * 32;    // batch tile
    const int iBase = blockIdx.y * 32;    // node tile

#pragma unroll
    for (int j = 0; j < 4; ++j) {
        int i = ty * 4 + j;
        tile[i][tx] = ws[(iBase + i) * BATCH + bBase + tx];   // coalesced read
    }
    __syncthreads();
#pragma unroll
    for (int j = 0; j < 4; ++j) {
        int r = ty * 4 + j;
        out[(bBase + r) * NNODES + iBase + tx] = tile[tx][r]; // coalesced write
    }
}

// ---------------------------------------------------------------------------
extern "C" void kernel_launch(void* const* d_in, const int* in_sizes, int n_in,
                              void* d_out, int out_size, void* d_ws, size_t ws_size,
                              hipStream_t stream) {
    const float* z       = (const float*)d_in[0];   // (B, L)
    const float* weights = (const float*)d_in[1];   // (N, 145)
    // d_in[2] = parent_mask (unused: weights are pre-masked)
    const int*   parents = (const int*)d_in[3];     // (N, K)
    float* out = (float*)d_out;                     // (B, N)
    float* ws  = (float*)d_ws;                      // node-major (N, B) scratch, 32 MB

    // Phase 1: WMMA GEMM for latent + bias, node-major into ws
    {
        const int waves = (BATCH / 16) * (NNODES / 16);   // 32768
        dim3 grid(waves / 8), block(256);
        wmma_pre_kernel<<<grid, block, 0, stream>>>(z, weights, ws);
    }
    // Phase 2: sequential DAG propagation over nodes, parallel over batch
    {
        dim3 grid(BATCH / 256), block(256);
        dag_sweep_kernel<<<grid, block, 0, stream>>>(weights, parents, ws);
    }
    // Phase 3: transpose to (B, N)
    {
        dim3 grid(BATCH / 32, NNODES / 32), block(32, 8);
        transpose_kernel<<<grid, block, 0, stream>>>(ws, out);
    }
}